// RadiomicsPreservationLoss_78417512890772
// MI455X (gfx1250) — compile-verified
//
#include <hip/hip_runtime.h>
#include <stdint.h>

typedef float v2f __attribute__((ext_vector_type(2)));
typedef float v8f __attribute__((ext_vector_type(8)));

#define W_ 512
#define H_ 512
#define TILE_H 8
#define LSTRIDE 520            // floats per LDS row (512 data + halo + align pad)
#define EPSF 1e-8f

// ---------------------------------------------------------------------------
// CDNA5 async global->LDS copy (ASYNCcnt-tracked, 16B per lane)
// ---------------------------------------------------------------------------
__device__ __forceinline__ void async_ld_b128(uint32_t lds_byte_off, const float* g) {
  asm volatile("global_load_async_to_lds_b128 %0, %1, off"
               :: "v"(lds_byte_off), "v"((unsigned long long)(uintptr_t)g)
               : "memory");
}
__device__ __forceinline__ void wait_async0() {
  asm volatile("s_wait_asynccnt 0" ::: "memory");
}

// ---------------------------------------------------------------------------
// Kernel 1: zero the 11 global accumulators in workspace
// ---------------------------------------------------------------------------
__global__ void radiomics_zero_ws(float* __restrict__ gsum) {
  if (threadIdx.x < 16) gsum[threadIdx.x] = 0.f;
}

// ---------------------------------------------------------------------------
// Kernel 2: per-strip partial sums
//   gsum[0]   = sum(mask)
//   gsum[1-4] = sum(in^k * mask), k=1..4
//   gsum[5-8] = sum(out^k * mask), k=1..4
//   gsum[9]   = sum |boxvar(out) - boxvar(in)| * mask
//   gsum[10]  = sum |lap(out)    - lap(in)   | * mask
// ---------------------------------------------------------------------------
__global__ __launch_bounds__(256) void radiomics_partials(
    const float* __restrict__ inp, const float* __restrict__ outp,
    const float* __restrict__ msk, float* __restrict__ gsum)
{
  __shared__ float sI[10 * LSTRIDE];
  __shared__ float sO[10 * LSTRIDE];
  __shared__ float part[8][16];

  const int tid = threadIdx.x;
  const int blk = blockIdx.x;
  const int bb  = blk >> 6;              // batch index (64 strips / image)
  const int rs  = (blk & 63) * TILE_H;   // first row of this strip

  const uint32_t lbI = (uint32_t)(uintptr_t)&sI[0];
  const uint32_t lbO = (uint32_t)(uintptr_t)&sO[0];
  const float* gI = inp  + (size_t)bb * H_ * W_;
  const float* gO = outp + (size_t)bb * H_ * W_;

  // zero the left/right halo columns (row data lives at float index 4..515)
  if (tid < 10) {
    sI[tid * LSTRIDE + 3] = 0.f;  sI[tid * LSTRIDE + 516] = 0.f;
    sO[tid * LSTRIDE + 3] = 0.f;  sO[tid * LSTRIDE + 516] = 0.f;
  }

  // stage rows rs-1 .. rs+8 (10 rows x 512 floats) of in/out into LDS.
  // 1280 float4 chunks per array; async copy 16B/lane; OOB rows -> zeros.
  for (int c = tid; c < 1280; c += 256) {
    const int i    = c >> 7;       // tile row 0..9
    const int j    = c & 127;      // float4 index within row
    const int grow = rs - 1 + i;   // global image row
    const uint32_t off = (uint32_t)((i * LSTRIDE + 4 + (j << 2)) << 2);
    if (grow >= 0 && grow < H_) {
      async_ld_b128(lbI + off, gI + (size_t)grow * W_ + (j << 2));
      async_ld_b128(lbO + off, gO + (size_t)grow * W_ + (j << 2));
    } else {
      float4 z = make_float4(0.f, 0.f, 0.f, 0.f);
      *(float4*)&sI[i * LSTRIDE + 4 + (j << 2)] = z;
      *(float4*)&sO[i * LSTRIDE + 4 + (j << 2)] = z;
    }
  }
  wait_async0();        // this wave's async copies complete
  __syncthreads();      // all waves' copies visible

  float acc[11];
#pragma unroll
  for (int k = 0; k < 11; ++k) acc[k] = 0.f;

  // Thread mapping: 128 threads per 4-row group, 2 groups.
  //   half = tid>>7 selects rows {0..3} or {4..7} of the strip,
  //   cb   = (tid&127)*4 is a 16B-aligned 4-column block.
  // Sliding 3x6 register window over 4 consecutive rows: each step reloads
  // only the new bottom row (1x ds_load_b128 + 2 scalar halo loads / image).
  const int half = tid >> 7;
  const int cb   = (tid & 127) << 2;
  const int tr0  = half * 4;                       // top tile row of window
  const float inv9 = 1.f / 9.f;

  float xi0[6], xi1[6], xi2[6], xo0[6], xo1[6], xo2[6];

  // load one 6-wide window row from LDS (b128 middle + 2 halo scalars)
  #define LOAD_ROW(dst, s, ti)                                            \
    {                                                                     \
      const int _b = (ti) * LSTRIDE + 4 + cb;                             \
      float4 _v = *(const float4*)&s[_b];                                 \
      dst[0] = s[_b - 1]; dst[1] = _v.x; dst[2] = _v.y;                   \
      dst[3] = _v.z;      dst[4] = _v.w; dst[5] = s[_b + 4];              \
    }

  LOAD_ROW(xi0, sI, tr0 + 0)  LOAD_ROW(xi1, sI, tr0 + 1)
  LOAD_ROW(xo0, sO, tr0 + 0)  LOAD_ROW(xo1, sO, tr0 + 1)

  const float* mrow0 = msk + ((size_t)bb * H_ + rs + tr0) * W_ + cb;

#pragma unroll
  for (int r = 0; r < 4; ++r) {
    LOAD_ROW(xi2, sI, tr0 + r + 2)
    LOAD_ROW(xo2, sO, tr0 + r + 2)
    const float4 mk = *(const float4*)(mrow0 + (size_t)r * W_);
    const float mv[4] = {mk.x, mk.y, mk.z, mk.w};

    // column sums and column sums-of-squares, shared by the 4 pixels
    float csi[6], cqi[6], cso[6], cqo[6];
#pragma unroll
    for (int j = 0; j < 6; ++j) {
      csi[j] = xi0[j] + xi1[j] + xi2[j];
      cqi[j] = xi0[j] * xi0[j] + xi1[j] * xi1[j] + xi2[j] * xi2[j];
      cso[j] = xo0[j] + xo1[j] + xo2[j];
      cqo[j] = xo0[j] * xo0[j] + xo1[j] * xo1[j] + xo2[j] * xo2[j];
    }

#pragma unroll
    for (int p = 0; p < 4; ++p) {                  // 4 pixels; center col p+1
      const float m = mv[p];
      const float sxi  = csi[p] + csi[p + 1] + csi[p + 2];
      const float sxxi = cqi[p] + cqi[p + 1] + cqi[p + 2];
      const float sxo  = cso[p] + cso[p + 1] + cso[p + 2];
      const float sxxo = cqo[p] + cqo[p + 1] + cqo[p + 2];
      const float mi  = sxi * inv9, mo = sxo * inv9;
      const float itv = sxxi * inv9 - mi * mi;     // local variance (in)
      const float otv = sxxo * inv9 - mo * mo;     // local variance (out)
      const float lapi = xi0[p + 1] + xi2[p + 1] + xi1[p] + xi1[p + 2]
                         - 4.f * xi1[p + 1];
      const float lapo = xo0[p + 1] + xo2[p + 1] + xo1[p] + xo1[p + 2]
                         - 4.f * xo1[p + 1];
      acc[9]  += m * fabsf(otv - itv);
      acc[10] += m * fabsf(lapo - lapi);

      const float a = xi1[p + 1], b = xo1[p + 1];
      acc[0] += m;
      float am = a * m; acc[1] += am; am *= a; acc[2] += am;
      am *= a; acc[3] += am; am *= a; acc[4] += am;
      float bm = b * m; acc[5] += bm; bm *= b; acc[6] += bm;
      bm *= b; acc[7] += bm; bm *= b; acc[8] += bm;
    }

    // slide the window down one row
#pragma unroll
    for (int j = 0; j < 6; ++j) {
      xi0[j] = xi1[j]; xi1[j] = xi2[j];
      xo0[j] = xo1[j]; xo1[j] = xo2[j];
    }
  }
  #undef LOAD_ROW

  // ---- wave32 reduction of the 11 accumulators ----
#pragma unroll
  for (int k = 0; k < 11; ++k) {
    float v = acc[k];
    v += __shfl_down(v, 16, 32);
    v += __shfl_down(v, 8, 32);
    v += __shfl_down(v, 4, 32);
    v += __shfl_down(v, 2, 32);
    v += __shfl_down(v, 1, 32);
    acc[k] = v;
  }
  const int lane = tid & 31, wv = tid >> 5;
  if (lane == 0) {
#pragma unroll
    for (int k = 0; k < 16; ++k) part[wv][k] = (k < 11) ? acc[k] : 0.f;
  }
  __syncthreads();

  // ---- cross-wave reduction (8 waves x 16 slots) via V_WMMA_F32_16X16X4_F32:
  // D = ones(16x4) x B(4x16) + C  =>  D[:,n] = sum over K of B[k][n].
  // Column n of B occupies lanes {n, n+16} x 2 VGPRs -> 4 wave-partials per
  // WMMA; two chained WMMAs sum all 8 waves; row 0 of D holds the totals.
  if (tid < 32) {
    const int n16 = tid & 15, h = tid >> 4;
    v2f a;  a.x = 1.f;               a.y = 1.f;
    v2f b0; b0.x = part[h][n16];     b0.y = part[2 + h][n16];
    v2f b1; b1.x = part[4 + h][n16]; b1.y = part[6 + h][n16];
    v8f cacc = {0.f, 0.f, 0.f, 0.f, 0.f, 0.f, 0.f, 0.f};
    cacc = __builtin_amdgcn_wmma_f32_16x16x4_f32(false, a, false, b0,
                                                 (short)0, cacc, false, false);
    cacc = __builtin_amdgcn_wmma_f32_16x16x4_f32(false, a, false, b1,
                                                 (short)0, cacc, false, false);
    if (tid < 11) atomicAdd(&gsum[tid], cacc[0]);
  }
}

// ---------------------------------------------------------------------------
// Kernel 3: raw moments -> central moments -> the 4 loss scalars
// ---------------------------------------------------------------------------
__global__ void radiomics_finalize(const float* __restrict__ gsum,
                                   float* __restrict__ o, float invN) {
  if (threadIdx.x != 0) return;
  const float Sm  = gsum[0];
  const float Si1 = gsum[1], Si2 = gsum[2], Si3 = gsum[3], Si4 = gsum[4];
  const float So1 = gsum[5], So2 = gsum[6], So3 = gsum[7], So4 = gsum[8];
  const float ms = Sm + EPSF;
  const float im = Si1 / ms, om = So1 / ms;
  const float im2 = im * im, om2 = om * om;

  const float c2i = Si2 - 2.f * im * Si1 + im2 * Sm;
  const float c3i = Si3 - 3.f * im * Si2 + 3.f * im2 * Si1 - im2 * im * Sm;
  const float c4i = Si4 - 4.f * im * Si3 + 6.f * im2 * Si2
                    - 4.f * im2 * im * Si1 + im2 * im2 * Sm;
  const float c2o = So2 - 2.f * om * So1 + om2 * Sm;
  const float c3o = So3 - 3.f * om * So2 + 3.f * om2 * So1 - om2 * om * Sm;
  const float c4o = So4 - 4.f * om * So3 + 6.f * om2 * So2
                    - 4.f * om2 * om * So1 + om2 * om2 * Sm;

  const float iv = c2i / ms, ov = c2o / ms;
  const float isk = c3i / (ms * (iv * sqrtf(iv) + EPSF));
  const float osk = c3o / (ms * (ov * sqrtf(ov) + EPSF));
  const float iku = c4i / (ms * (iv * iv + EPSF));
  const float oku = c4o / (ms * (ov * ov + EPSF));

  const float dm = im - om, dv = iv - ov, dsk = isk - osk, dku = iku - oku;
  const float intensity = dm * dm + dv * dv + dsk * dsk + dku * dku;
  const float texture = gsum[9] * invN;
  const float shape   = gsum[10] * invN;
  o[0] = intensity;
  o[1] = texture;
  o[2] = shape;
  o[3] = intensity + 0.5f * texture + shape;   // w = (1, 0.5, 1)
}

// ---------------------------------------------------------------------------
extern "C" void kernel_launch(void* const* d_in, const int* in_sizes, int n_in,
                              void* d_out, int out_size, void* d_ws, size_t ws_size,
                              hipStream_t stream) {
  const float* inp  = (const float*)d_in[0];
  const float* outp = (const float*)d_in[1];
  const float* msk  = (const float*)d_in[2];
  float* gsum = (float*)d_ws;
  float* o    = (float*)d_out;

  const int n = in_sizes[0];                 // 32*1*512*512
  const int batches = n / (H_ * W_);
  const int strips  = H_ / TILE_H;           // 64

  radiomics_zero_ws<<<1, 32, 0, stream>>>(gsum);
  radiomics_partials<<<batches * strips, 256, 0, stream>>>(inp, outp, msk, gsum);
  radiomics_finalize<<<1, 32, 0, stream>>>(gsum, o, 1.f / (float)n);
}